// ResidualVectorQuantizerEMA_17171279249687
// MI455X (gfx1250) — compile-verified
//
#include <hip/hip_runtime.h>
#include <stdint.h>

typedef __attribute__((ext_vector_type(2))) float v2f;
typedef __attribute__((ext_vector_type(4))) float v4f;
typedef __attribute__((ext_vector_type(8))) float v8f;

#define N_ROWS   65536
#define DIMD     256
#define KCODES   1024
#define LEVELS   8
#define TM       128      // rows per block (8 row tiles, one per wave)
#define CHUNK    64       // codebook rows staged per LDS buffer
#define NCHUNKS  (KCODES / CHUNK)          // 16 per level
#define GCHUNKS  (LEVELS * NCHUNKS)        // 128 total (codebook is contiguous)
#define LDSTRIDE 260      // padded row stride in floats (bank shift 4/row)

// ---------------------------------------------------------------------------
// Kernel 1: h[l][k] = 0.5 * ||emb[l][k]||^2
// ---------------------------------------------------------------------------
__global__ __launch_bounds__(256)
void rvq_norms(const float* __restrict__ emb, float* __restrict__ h) {
  int row = blockIdx.x * 256 + threadIdx.x;      // 0 .. L*K-1
  const float* e = emb + (size_t)row * DIMD;
  float s = 0.f;
  #pragma unroll 8
  for (int i = 0; i < DIMD / 4; ++i) {
    v4f v = *(const v4f*)(e + i * 4);
    s += v[0] * v[0] + v[1] * v[1] + v[2] * v[2] + v[3] * v[3];
  }
  h[row] = 0.5f * s;
}

// ---------------------------------------------------------------------------
// Async stage of one 64-code chunk (64 KB) into an LDS buffer.
// Each of the 256 threads issues 16 async b128 copies (ASYNCcnt-tracked).
// ---------------------------------------------------------------------------
__device__ __forceinline__ void async_fill(const float* __restrict__ src,
                                           float* __restrict__ ebuf, int tid) {
  #pragma unroll
  for (int i = 0; i < 16; ++i) {
    int f  = i * 256 + tid;          // float4 index within 64x256 chunk
    int r  = f >> 6;                 // code row
    int c4 = f & 63;
    unsigned lds = (unsigned)(uintptr_t)(ebuf + r * LDSTRIDE + c4 * 4);
    unsigned long long ga = (unsigned long long)(uintptr_t)(src + f * 4);
    asm volatile("global_load_async_to_lds_b128 %0, %1, off"
                 :: "v"(lds), "v"(ga) : "memory");
  }
}

__device__ __forceinline__ void wait_async0() {
  asm volatile("s_wait_asynccnt 0x0" ::: "memory");
}

// ---------------------------------------------------------------------------
// Kernel 2: full RVQ forward. One block owns TM=128 rows for all 8 levels.
// Residual tile stays in LDS across levels; codebook double-buffered via
// async-to-LDS copies; distances via v_wmma_f32_16x16x4_f32 (exact f32).
// ---------------------------------------------------------------------------
__global__ __launch_bounds__(256)
void rvq_main(const float* __restrict__ x,
              const float* __restrict__ emb,
              const float* __restrict__ h,
              float* __restrict__ q_out,
              float* __restrict__ loss_out,
              float* __restrict__ idx_out) {
  extern __shared__ __align__(16) char smem_raw[];
  float* R       = (float*)smem_raw;                 // TM x LDSTRIDE (residual)
  float* EB0     = R + TM * LDSTRIDE;                // CHUNK x LDSTRIDE (ping)
  float* EB1     = EB0 + CHUNK * LDSTRIDE;           // CHUNK x LDSTRIDE (pong)
  int*   tileIdx = (int*)(EB1 + CHUNK * LDSTRIDE);   // TM
  float* lred    = (float*)(tileIdx + TM);           // 256

  const int tid  = threadIdx.x;
  const int wid  = tid >> 5;        // wave id == row tile (0..7)
  const int lane = tid & 31;
  const int half = lane >> 4;       // which 16-lane half
  const int l16  = lane & 15;
  const int row0 = blockIdx.x * TM;

  // ---- load x tile into LDS as initial residual (coalesced b128) ----
  #pragma unroll 4
  for (int i = 0; i < 32; ++i) {
    int f  = i * 256 + tid;         // float4 index within 128x256 tile
    int r  = f >> 6;
    int c4 = f & 63;
    v4f v = *(const v4f*)(x + (size_t)(row0 + r) * DIMD + c4 * 4);
    *(v4f*)(R + r * LDSTRIDE + c4 * 4) = v;
  }

  // ---- prefetch first codebook chunk ----
  async_fill(emb, EB0, tid);

  float losspart = 0.f;

  for (int l = 0; l < LEVELS; ++l) {
    const float* embL = emb + (size_t)l * KCODES * DIMD;
    const float* hL   = h + l * KCODES;

    float rmin[8];
    int   ridx[8];
    #pragma unroll
    for (int j = 0; j < 8; ++j) { rmin[j] = __builtin_inff(); ridx[j] = 0; }

    for (int chunk = 0; chunk < NCHUNKS; ++chunk) {
      const int gc = l * NCHUNKS + chunk;            // flat chunk id 0..127
      wait_async0();         // this wave's fill of buf[gc&1] has landed
      __syncthreads();       // all waves' fills landed; prior compute/update done

      // prefetch next chunk into the other buffer (safe: its last consumer
      // finished before the barrier above)
      if (gc + 1 < GCHUNKS)
        async_fill(emb + (size_t)(gc + 1) * CHUNK * DIMD,
                   ((gc + 1) & 1) ? EB1 : EB0, tid);

      float* EB = (gc & 1) ? EB1 : EB0;

      // ---- four 16x16 output tiles via f32 WMMA; A shared per K-step ----
      v8f acc0 = {}, acc1 = {}, acc2 = {}, acc3 = {};
      // A layout (16x4 f32): lane<16 -> M=lane, K=kk..kk+1; lane>=16 -> K=kk+2..kk+3
      const float* Arow = R  + (wid * 16 + l16) * LDSTRIDE + 2 * half;
      const float* B0   = EB + ( 0 + l16) * LDSTRIDE + 2 * half;
      const float* B1   = EB + (16 + l16) * LDSTRIDE + 2 * half;
      const float* B2   = EB + (32 + l16) * LDSTRIDE + 2 * half;
      const float* B3   = EB + (48 + l16) * LDSTRIDE + 2 * half;
      #pragma unroll 2
      for (int kk = 0; kk < DIMD; kk += 4) {
        v2f a  = *(const v2f*)(Arow + kk);
        v2f b0 = *(const v2f*)(B0 + kk);
        v2f b1 = *(const v2f*)(B1 + kk);
        v2f b2 = *(const v2f*)(B2 + kk);
        v2f b3 = *(const v2f*)(B3 + kk);
        acc0 = __builtin_amdgcn_wmma_f32_16x16x4_f32(
            false, a, false, b0, (short)0, acc0, false, false);
        acc1 = __builtin_amdgcn_wmma_f32_16x16x4_f32(
            false, a, false, b1, (short)0, acc1, false, false);
        acc2 = __builtin_amdgcn_wmma_f32_16x16x4_f32(
            false, a, false, b2, (short)0, acc2, false, false);
        acc3 = __builtin_amdgcn_wmma_f32_16x16x4_f32(
            false, a, false, b3, (short)0, acc3, false, false);
      }

      // ---- running argmin of (h[col] - dot), ascending col = tie->low idx ----
      int col0 = chunk * CHUNK + l16;
      float h0 = hL[col0];
      float h1 = hL[col0 + 16];
      float h2 = hL[col0 + 32];
      float h3 = hL[col0 + 48];
      #pragma unroll
      for (int j = 0; j < 8; ++j) {
        float m0 = h0 - acc0[j];
        if (m0 < rmin[j]) { rmin[j] = m0; ridx[j] = col0; }
        float m1 = h1 - acc1[j];
        if (m1 < rmin[j]) { rmin[j] = m1; ridx[j] = col0 + 16; }
        float m2 = h2 - acc2[j];
        if (m2 < rmin[j]) { rmin[j] = m2; ridx[j] = col0 + 32; }
        float m3 = h3 - acc3[j];
        if (m3 < rmin[j]) { rmin[j] = m3; ridx[j] = col0 + 48; }
      }
    }

    // ---- cross-lane argmin within each 16-lane half (C layout: row j / j+8) ----
    #pragma unroll
    for (int j = 0; j < 8; ++j) {
      float v = rmin[j];
      int   ix = ridx[j];
      #pragma unroll
      for (int off = 1; off < 16; off <<= 1) {
        float ov = __shfl_xor(v, off, 32);
        int   oi = __shfl_xor(ix, off, 32);
        if (ov < v || (ov == v && oi < ix)) { v = ov; ix = oi; }
      }
      if (l16 == 0) {
        int r = wid * 16 + half * 8 + j;   // this wave owns complete rows
        tileIdx[r] = ix;
        idx_out[(size_t)(row0 + r) * LEVELS + l] = (float)ix;
      }
    }
    __syncthreads();   // tileIdx visible; all waves done reading R for WMMA

    // ---- residual update r -= e[idx]; loss += ||r_new||^2 ----
    {
      int r    = tid >> 1;
      int cseg = (tid & 1) * 128;
      const float* e = embL + (size_t)tileIdx[r] * DIMD + cseg;
      float* Rr = R + r * LDSTRIDE + cseg;
      #pragma unroll 4
      for (int i = 0; i < 32; ++i) {
        v4f ev = *(const v4f*)(e + i * 4);
        v4f rv = *(v4f*)(Rr + i * 4);
        rv -= ev;
        *(v4f*)(Rr + i * 4) = rv;
        losspart += rv[0] * rv[0] + rv[1] * rv[1] + rv[2] * rv[2] + rv[3] * rv[3];
      }
    }
    // next iteration's wait+barrier separates this update from WMMA reads
  }

  __syncthreads();
  // ---- q_out = x - r_final ----
  #pragma unroll 4
  for (int i = 0; i < 32; ++i) {
    int f  = i * 256 + tid;
    int r  = f >> 6;
    int c4 = f & 63;
    v4f xv = *(const v4f*)(x + (size_t)(row0 + r) * DIMD + c4 * 4);
    v4f rv = *(v4f*)(R + r * LDSTRIDE + c4 * 4);
    *(v4f*)(q_out + (size_t)(row0 + r) * DIMD + c4 * 4) = xv - rv;
  }

  // ---- block loss reduction + single atomic ----
  lred[tid] = losspart;
  __syncthreads();
  for (int s = 128; s > 0; s >>= 1) {
    if (tid < s) lred[tid] += lred[tid + s];
    __syncthreads();
  }
  if (tid == 0)
    atomicAdd(loss_out, lred[0] * (0.25f / ((float)N_ROWS * (float)DIMD)));
}

// ---------------------------------------------------------------------------
extern "C" void kernel_launch(void* const* d_in, const int* in_sizes, int n_in,
                              void* d_out, int out_size, void* d_ws, size_t ws_size,
                              hipStream_t stream) {
  const float* x   = (const float*)d_in[0];
  const float* emb = (const float*)d_in[1];

  float* q_out   = (float*)d_out;                          // [N, D]
  float* loss    = q_out + (size_t)N_ROWS * DIMD;          // scalar
  float* idx_out = loss + 1;                               // [N, L] as float
  float* h       = (float*)d_ws;                           // [L, K]

  hipMemsetAsync(loss, 0, sizeof(float), stream);

  rvq_norms<<<(LEVELS * KCODES) / 256, 256, 0, stream>>>(emb, h);

  size_t smem = (size_t)(TM + 2 * CHUNK) * LDSTRIDE * sizeof(float)  // R + EB0/1
              + TM * sizeof(int)                                      // tileIdx
              + 256 * sizeof(float);                                  // lred
  rvq_main<<<N_ROWS / TM, 256, smem, stream>>>(x, emb, h, q_out, loss, idx_out);
}